// NodeBlock_2740189135078
// MI455X (gfx1250) — compile-verified
//
#include <hip/hip_runtime.h>

typedef __attribute__((ext_vector_type(2))) float v2f;
typedef __attribute__((ext_vector_type(8))) float v8f;

#define DE   32
#define DN   64
#define DC   32
#define DIN  160   // 2*DE + DN + DC
#define HID  256
#define DOUT 64
#define MROWS 32   // rows per wave: two 16-row M-tiles share each B load
#define VSTRIDE 164   // 160+4 pad: lane m -> bank 36m (mod 64), all distinct
#define HSTRIDE 260   // 256+4 pad: lane m -> bank 4m  (mod 64), all distinct

// ---------------- K1: zero accumulator workspace ----------------
__global__ void zero_ws_kernel(float* __restrict__ ws, int n) {
  int i = blockIdx.x * blockDim.x + threadIdx.x;
  if (i < n) ws[i] = 0.0f;
}

// ---------------- K1b: pack weights into k-pair interleaved layout ----------
// Wp[(p*C + c)*2 + {0,1}] = W[(2p)*C + c], W[(2p+1)*C + c]
__global__ void pack_pairs_kernel(const float* __restrict__ W,
                                  float* __restrict__ Wp, int K2, int C) {
  int i = blockIdx.x * blockDim.x + threadIdx.x;
  if (i >= K2 * C) return;
  int p = i / C, c = i - p * C;
  Wp[2 * i]     = W[(size_t)(2 * p) * C + c];
  Wp[2 * i + 1] = W[(size_t)(2 * p + 1) * C + c];
}

// ---------------- K2: edge -> node scatter (sums + counts) ----------------
__global__ __launch_bounds__(256) void scatter_edges_kernel(
    const float* __restrict__ edge_feat,
    const int*   __restrict__ receivers,
    const int*   __restrict__ senders,
    float* __restrict__ in_sum,  float* __restrict__ out_sum,
    float* __restrict__ in_cnt,  float* __restrict__ out_cnt,
    int E) {
  int gid = blockIdx.x * blockDim.x + threadIdx.x;
  int e = gid >> 5;          // edge index
  int c = gid & 31;          // feature index (coalesced)
  if (e >= E) return;
  float v = edge_feat[(size_t)e * DE + c];
  int r = receivers[e];
  int s = senders[e];
  atomicAdd(&in_sum[(size_t)r * DE + c], v);
  atomicAdd(&out_sum[(size_t)s * DE + c], v);
  if (c == 0) {
    atomicAdd(&in_cnt[r], 1.0f);
    atomicAdd(&out_cnt[s], 1.0f);
  }
}

// ------- K3: 32 rows per wave, MLP via V_WMMA_F32_16X16X4_F32 -------
__global__ __launch_bounds__(32) void node_mlp_kernel(
    const float* __restrict__ in_sum,  const float* __restrict__ out_sum,
    const float* __restrict__ in_cnt,  const float* __restrict__ out_cnt,
    const float* __restrict__ node_feat, const float* __restrict__ ctx,
    const float* __restrict__ W1p, const float* __restrict__ b1,
    const float* __restrict__ W2p, const float* __restrict__ b2,
    float* __restrict__ out, int N) {
  __shared__ float lin[MROWS * VSTRIDE];   // vtin tile [32 x 160] (padded)
  __shared__ float lhid[MROWS * HSTRIDE];  // hidden tile [32 x 256] (padded)

  const int lane = threadIdx.x;            // wave32
  const int row0 = blockIdx.x * MROWS;

  // ---- build vtin tile in LDS: [in_agg | out_agg | node_feat | ctx] ----
  for (int idx = lane; idx < MROWS * DIN; idx += 32) {
    int r  = idx / DIN;
    int cc = idx - r * DIN;
    int node = row0 + r;
    if (node >= N) node = N - 1;  // clamp (loads only; stores are guarded)
    float v;
    if (cc < DE) {
      v = in_sum[(size_t)node * DE + cc] / fmaxf(in_cnt[node], 1.0f);
    } else if (cc < 2 * DE) {
      v = out_sum[(size_t)node * DE + (cc - DE)] / fmaxf(out_cnt[node], 1.0f);
    } else if (cc < 2 * DE + DN) {
      v = node_feat[(size_t)node * DN + (cc - 2 * DE)];
    } else {
      v = ctx[cc - 2 * DE - DN];
    }
    lin[r * VSTRIDE + cc] = v;
  }
  __syncthreads();

  const int m     = lane & 15;          // A row within tile / B,C,D column
  const int phalf = lane >> 4;          // which K-pair of the group of 4
  const int khalf = phalf << 1;         // K offset: lanes 16-31 hold K+2,K+3
  const int mhi   = phalf << 3;         // C/D: VGPR j holds M = j + 8*(lane/16)

  // ---- GEMM1: [32x160] @ W1[160x256] + b1, ReLU -> LDS hidden ----
  for (int nt = 0; nt < HID / 16; ++nt) {
    const int col = nt * 16 + m;
    v8f acc0 = {}, acc1 = {};
#pragma unroll 8
    for (int kk = 0; kk < DIN / 4; ++kk) {
      const int k = kk * 4 + khalf;
      const int p = kk * 2 + phalf;     // k-pair index (k == 2*p)
      v2f a0, a1;
      a0.x = lin[m * VSTRIDE + k];
      a0.y = lin[m * VSTRIDE + k + 1];
      a1.x = lin[(m + 16) * VSTRIDE + k];
      a1.y = lin[(m + 16) * VSTRIDE + k + 1];
      const v2f b = *(const v2f*)(W1p + ((size_t)p * HID + col) * 2);
      acc0 = __builtin_amdgcn_wmma_f32_16x16x4_f32(false, a0, false, b,
                                                   (short)0, acc0, false, false);
      acc1 = __builtin_amdgcn_wmma_f32_16x16x4_f32(false, a1, false, b,
                                                   (short)0, acc1, false, false);
    }
    const float bias = b1[col];
#pragma unroll
    for (int j = 0; j < 8; ++j) {
      lhid[(j + mhi) * HSTRIDE + col]      = fmaxf(acc0[j] + bias, 0.0f);
      lhid[(j + mhi + 16) * HSTRIDE + col] = fmaxf(acc1[j] + bias, 0.0f);
    }
  }
  __syncthreads();

  // ---- GEMM2: [32x256] @ W2[256x64] + b2 -> out ----
  for (int nt = 0; nt < DOUT / 16; ++nt) {
    const int col = nt * 16 + m;
    v8f acc0 = {}, acc1 = {};
#pragma unroll 8
    for (int kk = 0; kk < HID / 4; ++kk) {
      const int k = kk * 4 + khalf;
      const int p = kk * 2 + phalf;
      v2f a0, a1;
      a0.x = lhid[m * HSTRIDE + k];
      a0.y = lhid[m * HSTRIDE + k + 1];
      a1.x = lhid[(m + 16) * HSTRIDE + k];
      a1.y = lhid[(m + 16) * HSTRIDE + k + 1];
      const v2f b = *(const v2f*)(W2p + ((size_t)p * DOUT + col) * 2);
      acc0 = __builtin_amdgcn_wmma_f32_16x16x4_f32(false, a0, false, b,
                                                   (short)0, acc0, false, false);
      acc1 = __builtin_amdgcn_wmma_f32_16x16x4_f32(false, a1, false, b,
                                                   (short)0, acc1, false, false);
    }
    const float bias = b2[col];
#pragma unroll
    for (int j = 0; j < 8; ++j) {
      const int rA = row0 + j + mhi;
      const int rB = rA + 16;
      if (rA < N) out[(size_t)rA * DOUT + col] = acc0[j] + bias;
      if (rB < N) out[(size_t)rB * DOUT + col] = acc1[j] + bias;
    }
  }
}

extern "C" void kernel_launch(void* const* d_in, const int* in_sizes, int n_in,
                              void* d_out, int out_size, void* d_ws, size_t ws_size,
                              hipStream_t stream) {
  const float* node_feat = (const float*)d_in[0];
  const float* edge_feat = (const float*)d_in[1];
  const float* ctx       = (const float*)d_in[2];
  const float* W1        = (const float*)d_in[3];
  const float* b1        = (const float*)d_in[4];
  const float* W2        = (const float*)d_in[5];
  const float* b2        = (const float*)d_in[6];
  const int*   receivers = (const int*)d_in[7];
  const int*   senders   = (const int*)d_in[8];

  const int N = in_sizes[0] / DN;   // 50000
  const int E = in_sizes[1] / DE;   // 1600000

  // workspace: in_sum[N*32] | out_sum[N*32] | in_cnt[N] | out_cnt[N] | W1p | W2p
  float* ws      = (float*)d_ws;
  float* in_sum  = ws;
  float* out_sum = ws + (size_t)N * DE;
  float* in_cnt  = ws + (size_t)2 * N * DE;
  float* out_cnt = in_cnt + N;
  float* W1p     = out_cnt + N;               // DIN*HID floats
  float* W2p     = W1p + (size_t)DIN * HID;   // HID*DOUT floats

  const int nzero = N * (2 * DE + 2);
  zero_ws_kernel<<<(nzero + 255) / 256, 256, 0, stream>>>(ws, nzero);

  const int n1 = (DIN / 2) * HID;
  pack_pairs_kernel<<<(n1 + 255) / 256, 256, 0, stream>>>(W1, W1p, DIN / 2, HID);
  const int n2 = (HID / 2) * DOUT;
  pack_pairs_kernel<<<(n2 + 255) / 256, 256, 0, stream>>>(W2, W2p, HID / 2, DOUT);

  const long long tot = (long long)E * DE;
  scatter_edges_kernel<<<(int)((tot + 255) / 256), 256, 0, stream>>>(
      edge_feat, receivers, senders, in_sum, out_sum, in_cnt, out_cnt, E);

  const int tiles = (N + MROWS - 1) / MROWS;
  node_mlp_kernel<<<tiles, 32, 0, stream>>>(
      in_sum, out_sum, in_cnt, out_cnt, node_feat, ctx,
      W1p, b1, W2p, b2, (float*)d_out, N);
}